// NaMixedOp_40836549050697
// MI455X (gfx1250) — compile-verified
//
#include <hip/hip_runtime.h>

#define DIN 64

typedef __attribute__((ext_vector_type(2))) float v2f;
typedef __attribute__((ext_vector_type(8))) float v8f;

// ---------------------------------------------------------------------------
// 1) zero the scratch region (degrees + 4 aggregate arrays)
// ---------------------------------------------------------------------------
__global__ void zero_ws_kernel(float* __restrict__ p, int n) {
    int i = blockIdx.x * blockDim.x + threadIdx.x;
    if (i < n) p[i] = 0.0f;
}

// ---------------------------------------------------------------------------
// 2) weighted + count in-degree via f32 atomics (L2 atomic units)
// ---------------------------------------------------------------------------
__global__ void degree_kernel(const float* __restrict__ ew,
                              const int* __restrict__ dst,
                              float* __restrict__ deg_w,
                              float* __restrict__ deg_c, int E) {
    int e = blockIdx.x * blockDim.x + threadIdx.x;
    if (e < E) {
        int t = dst[e];
        unsafeAtomicAdd(deg_w + t, ew[e]);
        unsafeAtomicAdd(deg_c + t, 1.0f);
    }
}

// ---------------------------------------------------------------------------
// 3) deg_w = max(deg_w,1);  deg_c = 1/max(deg_c,1)  (stored as reciprocal)
// ---------------------------------------------------------------------------
__global__ void finalize_deg_kernel(float* __restrict__ deg_w,
                                    float* __restrict__ deg_c, int N) {
    int i = blockIdx.x * blockDim.x + threadIdx.x;
    if (i < N) {
        deg_w[i] = fmaxf(deg_w[i], 1.0f);
        deg_c[i] = 1.0f / fmaxf(deg_c[i], 1.0f);
    }
}

// ---------------------------------------------------------------------------
// 4) fold arch weights into 5 combined 64x64 matrices + one combined bias.
//    Wcomb is stored in WMMA-B-fragment-interleaved order so that each
//    B fragment (W[k][n], W[k+1][n]) is one contiguous 64-bit load:
//      k = 4*ks + 2*h + j,  addr = (((f*64 + ks*2 + h)*64 + n)<<1) + j
// ---------------------------------------------------------------------------
__global__ void combine_weights_kernel(
    const float* __restrict__ w, const float* __restrict__ eps,
    const float* __restrict__ Wgcn, const float* __restrict__ bgcn,
    const float* __restrict__ Wss,  const float* __restrict__ Wsn,
    const float* __restrict__ bs,
    const float* __restrict__ Wgin, const float* __restrict__ bgin,
    const float* __restrict__ Wgc1, const float* __restrict__ Wgc2,
    const float* __restrict__ bgc,
    const float* __restrict__ Wedge, const float* __restrict__ bedge,
    float* __restrict__ Wc, float* __restrict__ bias) {
    const int i = blockIdx.x * blockDim.x + threadIdx.x;
    if (i >= DIN * DIN) return;
    const float w0 = w[0], w1 = w[1], w2 = w[2], w3 = w[3], w4 = w[4];
    const float ge = 1.0f + eps[0];

    const int k = i >> 6, n = i & 63;
    const int ks = k >> 2, h = (k >> 1) & 1, j = k & 1;
    const int base = (((ks * 2 + h) * 64 + n) << 1) + j;  // within one feature

    Wc[0 * 8192 + base] = w0 * Wgcn[i];
    Wc[1 * 8192 + base] = w1 * Wss[i] + w2 * ge * Wgin[i] + w3 * Wgc1[i];
    Wc[2 * 8192 + base] = w1 * Wsn[i];
    Wc[3 * 8192 + base] = w2 * Wgin[i] + w3 * Wgc2[i];
    Wc[4 * 8192 + base] = w4 * Wedge[i];
    if (i < DIN)
        bias[i] = w0 * bgcn[i] + w1 * bs[i] + w2 * bgin[i] + w3 * bgc[i] +
                  w4 * bedge[i];
}

// ---------------------------------------------------------------------------
// 5) edge aggregation: 16 dim-quads x 16 edges per 256-thread block.
//    float4 (B128) loads of x[src] / edge_attr; 16 global_atomic_add_f32
//    per thread; the 51MB aggregate set is resident in the 192MB L2.
// ---------------------------------------------------------------------------
__global__ void __launch_bounds__(256)
edge_aggregate_kernel(const float* __restrict__ x,
                      const float* __restrict__ ew,
                      const float* __restrict__ eattr,
                      const int* __restrict__ src,
                      const int* __restrict__ dst,
                      const float* __restrict__ deg_w,
                      float* __restrict__ aggG, float* __restrict__ aggS,
                      float* __restrict__ aggW, float* __restrict__ aggE,
                      int E) {
    const int q = threadIdx.x;  // 0..15 dim quad
    const int e = blockIdx.x * blockDim.y + threadIdx.y;
    if (e >= E) return;
    const int s = src[e];
    const int t = dst[e];
    const float w = ew[e];
    const float norm = w * __frsqrt_rn(deg_w[s] * deg_w[t]);

    const float4 xs = ((const float4*)(x + (size_t)s * DIN))[q];
    const float4 ea = ((const float4*)(eattr + (size_t)e * DIN))[q];

    float4 m;
    m.x = fmaxf(xs.x + ea.x, 0.0f);
    m.y = fmaxf(xs.y + ea.y, 0.0f);
    m.z = fmaxf(xs.z + ea.z, 0.0f);
    m.w = fmaxf(xs.w + ea.w, 0.0f);

    float* pG = aggG + (size_t)t * DIN + q * 4;
    float* pS = aggS + (size_t)t * DIN + q * 4;
    float* pW = aggW + (size_t)t * DIN + q * 4;
    float* pE = aggE + (size_t)t * DIN + q * 4;

    unsafeAtomicAdd(pG + 0, norm * xs.x);
    unsafeAtomicAdd(pG + 1, norm * xs.y);
    unsafeAtomicAdd(pG + 2, norm * xs.z);
    unsafeAtomicAdd(pG + 3, norm * xs.w);

    unsafeAtomicAdd(pS + 0, xs.x);
    unsafeAtomicAdd(pS + 1, xs.y);
    unsafeAtomicAdd(pS + 2, xs.z);
    unsafeAtomicAdd(pS + 3, xs.w);

    unsafeAtomicAdd(pW + 0, w * xs.x);
    unsafeAtomicAdd(pW + 1, w * xs.y);
    unsafeAtomicAdd(pW + 2, w * xs.z);
    unsafeAtomicAdd(pW + 3, w * xs.w);

    unsafeAtomicAdd(pE + 0, w * m.x);
    unsafeAtomicAdd(pE + 1, w * m.y);
    unsafeAtomicAdd(pE + 2, w * m.z);
    unsafeAtomicAdd(pE + 3, w * m.w);
}

// ---------------------------------------------------------------------------
// 6) mixed GEMM: per wave, one 16-row x 64-col output tile.
//    A: global_load_b64 (direct global pointer per feature).
//    B: ds_load_b64 from the 160KB Wcomb staged once per block in LDS.
// ---------------------------------------------------------------------------
template <bool SCALED>
__device__ __forceinline__ void mma_feature(const float* __restrict__ F,
                                            const float* __restrict__ sWf,
                                            int rowA, int half, int l,
                                            float scale, v8f acc[4]) {
    const float* __restrict__ A = F + (size_t)rowA * DIN + 2 * half;
#pragma unroll
    for (int ks = 0; ks < 16; ++ks) {
        v2f a = *(const v2f*)(A + 4 * ks);
        if (SCALED) {
            a.x *= scale;
            a.y *= scale;
        }
        const float* __restrict__ brow = sWf + (((ks * 2 + half) * 64 + l) << 1);
#pragma unroll
        for (int nt = 0; nt < 4; ++nt) {
            v2f b = *(const v2f*)(brow + nt * 32);  // (W[k][n], W[k+1][n])
            acc[nt] = __builtin_amdgcn_wmma_f32_16x16x4_f32(
                false, a, false, b, (short)0, acc[nt], false, false);
        }
    }
}

__global__ void __launch_bounds__(256)
gemm_mix_kernel(const float* __restrict__ aggG, const float* __restrict__ x,
                const float* __restrict__ aggS, const float* __restrict__ aggW,
                const float* __restrict__ aggE,
                const float* __restrict__ invDegC,
                const float* __restrict__ Wc, const float* __restrict__ bias,
                float* __restrict__ out, int nTiles, int N) {
    extern __shared__ float sW[];  // 5 * 8192 floats = 160KB (of 320KB/WGP)

    // cooperative stage of all 5 B matrices (interleaved layout), float4
    {
        const float4* __restrict__ g4 = (const float4*)Wc;
        float4* __restrict__ s4 = (float4*)sW;
#pragma unroll
        for (int i = threadIdx.x; i < 5 * 2048; i += 256) s4[i] = g4[i];
    }
    __syncthreads();

    const int lane = threadIdx.x & 31;
    const int wave = threadIdx.x >> 5;
    const int tile = blockIdx.x * (blockDim.x >> 5) + wave;
    if (tile >= nTiles) return;  // wave-uniform: EXEC all-ones at every WMMA

    const int l = lane & 15;
    const int half = lane >> 4;
    const int row0 = tile << 4;
    int rowA = row0 + l;
    if (rowA >= N) rowA = N - 1;  // safe clamp (N%16==0 here anyway)

    // C init: bias depends only on column -> broadcast per accumulator
    v8f acc[4];
#pragma unroll
    for (int nt = 0; nt < 4; ++nt) {
        const float bc = bias[nt * 16 + l];
#pragma unroll
        for (int r = 0; r < 8; ++r) acc[nt][r] = bc;
    }

    const float sinv = invDegC[rowA];  // only SAGE-mean feature is scaled

    mma_feature<false>(aggG, sW + 0 * 8192, rowA, half, l, 1.0f, acc);
    mma_feature<false>(x,    sW + 1 * 8192, rowA, half, l, 1.0f, acc);
    mma_feature<true >(aggS, sW + 2 * 8192, rowA, half, l, sinv, acc);
    mma_feature<false>(aggW, sW + 3 * 8192, rowA, half, l, 1.0f, acc);
    mma_feature<false>(aggE, sW + 4 * 8192, rowA, half, l, 1.0f, acc);

#pragma unroll
    for (int nt = 0; nt < 4; ++nt) {
#pragma unroll
        for (int r = 0; r < 8; ++r) {
            const int row = row0 + r + 8 * half;
            if (row < N) out[(size_t)row * DIN + nt * 16 + l] = acc[nt][r];
        }
    }
}

// ---------------------------------------------------------------------------
extern "C" void kernel_launch(void* const* d_in, const int* in_sizes, int n_in,
                              void* d_out, int out_size, void* d_ws,
                              size_t ws_size, hipStream_t stream) {
    const float* x     = (const float*)d_in[0];
    const float* wK    = (const float*)d_in[1];
    const float* ew    = (const float*)d_in[2];
    const float* eattr = (const float*)d_in[3];
    const float* Wgcn  = (const float*)d_in[4];
    const float* bgcn  = (const float*)d_in[5];
    const float* Wss   = (const float*)d_in[6];
    const float* Wsn   = (const float*)d_in[7];
    const float* bs    = (const float*)d_in[8];
    const float* eps   = (const float*)d_in[9];
    const float* Wgin  = (const float*)d_in[10];
    const float* bgin  = (const float*)d_in[11];
    const float* Wgc1  = (const float*)d_in[12];
    const float* Wgc2  = (const float*)d_in[13];
    const float* bgc   = (const float*)d_in[14];
    const float* Wedge = (const float*)d_in[15];
    const float* bedge = (const float*)d_in[16];
    const int*   eidx  = (const int*)d_in[17];

    const int N = in_sizes[0] / DIN;
    const int E = in_sizes[2];
    const int* src = eidx;
    const int* dst = eidx + E;

    // ---- workspace layout (floats) ----
    float* ws    = (float*)d_ws;
    float* deg_w = ws;                        // [N]
    float* deg_c = deg_w + N;                 // [N] -> becomes 1/deg_c
    float* aggG  = deg_c + N;                 // [N,64]
    float* aggS  = aggG + (size_t)N * DIN;    // [N,64]
    float* aggW  = aggS + (size_t)N * DIN;    // [N,64]
    float* aggE  = aggW + (size_t)N * DIN;    // [N,64]
    float* Wc    = aggE + (size_t)N * DIN;    // [5,8192] interleaved B layout
    float* bias  = Wc + 5 * 8192;             // [64]

    // 1) zero degrees + aggregates (harness does not re-zero between replays)
    const int nzero = N * (2 + 4 * DIN);
    zero_ws_kernel<<<(nzero + 255) / 256, 256, 0, stream>>>(ws, nzero);

    // 2) in-degrees
    degree_kernel<<<(E + 255) / 256, 256, 0, stream>>>(ew, dst, deg_w, deg_c, E);

    // 3) clamp / invert
    finalize_deg_kernel<<<(N + 255) / 256, 256, 0, stream>>>(deg_w, deg_c, N);

    // 4) fold arch weights into combined matrices (B-fragment layout)
    combine_weights_kernel<<<(DIN * DIN + 255) / 256, 256, 0, stream>>>(
        wK, eps, Wgcn, bgcn, Wss, Wsn, bs, Wgin, bgin, Wgc1, Wgc2, bgc, Wedge,
        bedge, Wc, bias);

    // 5) edge aggregation (dominant, bandwidth/L2-atomic bound)
    dim3 eblk(16, 16);
    edge_aggregate_kernel<<<(E + 15) / 16, eblk, 0, stream>>>(
        x, ew, eattr, src, dst, deg_w, aggG, aggS, aggW, aggE, E);

    // 6) WMMA GEMM + mixture (160KB dynamic LDS per block)
    const int nTiles = (N + 15) / 16;
    gemm_mix_kernel<<<(nTiles + 7) / 8, 256, 5 * 8192 * sizeof(float), stream>>>(
        aggG, x, aggS, aggW, aggE, deg_c, Wc, bias, (float*)d_out, nTiles, N);
}